// AlignDRAWModel_412316861146
// MI455X (gfx1250) — compile-verified
//
#include <hip/hip_runtime.h>
#include <math.h>

// ---------------- model dims ----------------
#define BN     128          // batch
#define IMG_A  64
#define CCH    3
#define RN5    5
#define TSTEPS 32
#define LLEN   32
#define LINV   300
#define LHV    128
#define ZV     100
#define ENCV   512
#define DECV   512
#define ALV    512

typedef __attribute__((ext_vector_type(2))) float v2f;
typedef __attribute__((ext_vector_type(8))) float v8f;

// A joint segment of the K dimension: within it both the A-matrix and the
// W-matrix are contiguous row-major slabs.  C[m][n] += A(m,k)*W(n,k) summed
// over each segment independently (WMMA accumulation is additive over K).
struct JSeg {
    const float* a; int lda;
    const float* w; int ldw;
    int width;
};
struct GemmArgs {
    JSeg s[4];
    int ns;
};

#define NTILE 4   // 16-wide N tiles per wave (independent WMMA chains)

// ---------------------------------------------------------------------------
// GEMM: Cout[m][n] = sum_seg sum_k A(m,k)*W(n,k) + bias1[n]+bias2[n]+Cadd[m][n]
// fp32 WMMA 16x16x4.  One wave = FOUR 16x16 tiles (shared A fragment),
// 4 waves/block -> 256 output columns per block.
// A 16x4: lane%16 = M row; lanes<16 hold K={k,k+1}, lanes>=16 K={k+2,k+3}.
// B 4x16: lane%16 = N col; same K split across lane halves.
// C/D: VGPR vr: lanes0-15 -> M=vr, lanes16-31 -> M=vr+8; N = lane%16.
// Out-of-range rows/cols: clamp index + multiply-by-0 flag — the inner loop is
// fully straight-line (no EXEC churn, no branches), EXEC stays all-1s.
// ---------------------------------------------------------------------------
__global__ __launch_bounds__(128) void gemm_wmma_f32(
    GemmArgs G, const float* __restrict__ bias1, const float* __restrict__ bias2,
    const float* __restrict__ Cadd, int ldca, float* __restrict__ Cout, int ldc,
    int M, int N)
{
    const int lane = threadIdx.x & 31;
    const int wave = threadIdx.x >> 5;
    const int n0 = (blockIdx.x * 4 + wave) * (16 * NTILE);
    if (n0 >= N) return;
    const int m0 = blockIdx.y * 16;

    const int r15 = lane & 15;
    const int klo = (lane >> 4) << 1;          // 0 or 2

    int mr = m0 + r15;
    float mflag = 1.0f;
    if (mr >= M) { mr = M - 1; mflag = 0.0f; }

    int   nr[NTILE];
    float nflag[NTILE];
    #pragma unroll
    for (int t = 0; t < NTILE; ++t) {
        nr[t] = n0 + 16 * t + r15;
        nflag[t] = 1.0f;
        if (nr[t] >= N) { nr[t] = N - 1; nflag[t] = 0.0f; }
    }

    v8f acc[NTILE];
    #pragma unroll
    for (int t = 0; t < NTILE; ++t)
        #pragma unroll
        for (int i = 0; i < 8; ++i) acc[t][i] = 0.0f;

    for (int sgi = 0; sgi < G.ns; ++sgi) {
        const JSeg sg = G.s[sgi];
        const float* __restrict__ Ap = sg.a + (size_t)mr * sg.lda;
        const float* __restrict__ Wp[NTILE];
        #pragma unroll
        for (int t = 0; t < NTILE; ++t)
            Wp[t] = sg.w + (size_t)nr[t] * sg.ldw;
        const int Ws = sg.width;
        const int kq = Ws & ~3;

        for (int k = 0; k < kq; k += 4) {
            const int ka = k + klo;
            v2f a;
            a.x = Ap[ka]     * mflag;
            a.y = Ap[ka + 1] * mflag;
            #pragma unroll
            for (int t = 0; t < NTILE; ++t) {
                v2f b;
                b.x = Wp[t][ka]     * nflag[t];
                b.y = Wp[t][ka + 1] * nflag[t];
                acc[t] = __builtin_amdgcn_wmma_f32_16x16x4_f32(
                             false, a, false, b, (short)0, acc[t], false, false);
            }
        }
        const int rem = Ws - kq;                // 0..3, uniform
        if (rem) {
            const int ka  = kq + klo;
            const int i0  = (ka     < Ws) ? ka     : (Ws - 1);
            const int i1  = (ka + 1 < Ws) ? ka + 1 : (Ws - 1);
            const float f0 = (klo     < rem) ? 1.0f : 0.0f;
            const float f1 = (klo + 1 < rem) ? 1.0f : 0.0f;
            v2f a;
            a.x = Ap[i0] * (mflag * f0);
            a.y = Ap[i1] * (mflag * f1);
            #pragma unroll
            for (int t = 0; t < NTILE; ++t) {
                v2f b;
                b.x = Wp[t][i0] * (nflag[t] * f0);
                b.y = Wp[t][i1] * (nflag[t] * f1);
                acc[t] = __builtin_amdgcn_wmma_f32_16x16x4_f32(
                             false, a, false, b, (short)0, acc[t], false, false);
            }
        }
    }

    // ---- epilogue ----
    const int rbase = m0 + ((lane >> 4) << 3);
    #pragma unroll
    for (int t = 0; t < NTILE; ++t) {
        const int cc = n0 + 16 * t + r15;
        if (cc >= N) continue;
        float badd = 0.0f;
        if (bias1) badd += bias1[cc];
        if (bias2) badd += bias2[cc];
        #pragma unroll
        for (int vr = 0; vr < 8; ++vr) {
            const int rr = rbase + vr;
            if (rr < M) {
                float v = acc[t][vr] + badd;
                if (Cadd) v += Cadd[(size_t)rr * ldca + cc];
                Cout[(size_t)rr * ldc + cc] = v;
            }
        }
    }
}

// ---------------------------------------------------------------------------
// LSTM cell: gates (B,4H) chunked [i|f|g|o]; in-place safe on c.
// ---------------------------------------------------------------------------
__global__ void lstm_cell_k(const float* gates, const float* c_in,
                            float* h_out, float* c_out, int H,
                            float* hl, int hl_stride, int total)
{
    const int idx = blockIdx.x * blockDim.x + threadIdx.x;
    if (idx >= total) return;
    const int b = idx / H, n = idx % H;
    const float* g = gates + b * 4 * H;
    const float gi = g[n], gf = g[H + n], gg = g[2 * H + n], go = g[3 * H + n];
    const float si = 1.0f / (1.0f + expf(-gi));
    const float sf = 1.0f / (1.0f + expf(-gf));
    const float so = 1.0f / (1.0f + expf(-go));
    const float c2 = sf * c_in[idx] + si * tanhf(gg);
    const float h  = so * tanhf(c2);
    c_out[idx] = c2;
    h_out[idx] = h;
    if (hl) hl[b * hl_stride + n] = h;
}

// ---------------------------------------------------------------------------
// Attention window: p = h@attn_W.T + b -> Fx,Fy (B,5,64) normalized, gamma.
// One block (64 thr) per batch element.
// ---------------------------------------------------------------------------
__global__ void attn_k(const float* hdec, const float* attn_W, const float* attn_b,
                       float* Fx, float* Fy, float* gamma, int N)
{
    const int b = blockIdx.x, t = threadIdx.x;
    __shared__ float p[5];
    __shared__ float prm[4];                 // gx, gy, sigma2, delta
    __shared__ float Fxs[5 * 64], Fys[5 * 64];
    __shared__ float sums[10];

    if (t < 32) {
        for (int i = 0; i < 5; ++i) {
            float acc = 0.0f;
            for (int k = t; k < DECV; k += 32)
                acc += hdec[b * DECV + k] * attn_W[i * DECV + k];
            for (int off = 16; off; off >>= 1) acc += __shfl_xor(acc, off, 32);
            if (t == 0) p[i] = acc + attn_b[i];
        }
    }
    __syncthreads();
    if (t == 0) {
        prm[0] = 0.5f * (IMG_A + 1) * (p[0] + 1.0f);                     // gx
        prm[1] = 0.5f * (IMG_A + 1) * (p[1] + 1.0f);                     // gy
        prm[2] = expf(p[2]);                                             // sigma2
        prm[3] = ((float)(IMG_A - 1) / (float)(N - 1)) * expf(p[3]);     // delta
        gamma[b] = expf(p[4]);
    }
    __syncthreads();
    if (t < 64) {
        const float inv2s2 = 1.0f / (2.0f * prm[2]);
        for (int i = 0; i < N; ++i) {
            const float off = ((float)i - (float)N * 0.5f - 0.5f) * prm[3];
            const float dx = (float)t - (prm[0] + off);
            const float dy = (float)t - (prm[1] + off);
            Fxs[i * 64 + t] = expf(-dx * dx * inv2s2);
            Fys[i * 64 + t] = expf(-dy * dy * inv2s2);
        }
    }
    __syncthreads();
    if (t < 5) {
        float s = 0.0f;
        for (int a = 0; a < 64; ++a) s += Fxs[t * 64 + a];
        sums[t] = s;
    } else if (t >= 8 && t < 13) {
        const int i = t - 8;
        float s = 0.0f;
        for (int a = 0; a < 64; ++a) s += Fys[i * 64 + a];
        sums[5 + i] = s;
    }
    __syncthreads();
    if (t < 64) {
        for (int i = 0; i < N; ++i) {
            Fx[b * 320 + i * 64 + t] = Fxs[i * 64 + t] / (sums[i] + 1e-8f);
            Fy[b * 320 + i * 64 + t] = Fys[i * 64 + t] / (sums[5 + i] + 1e-8f);
        }
    }
}

// ---------------------------------------------------------------------------
// Glimpse: r[b] = gamma*[ Fy@img@Fx.T | Fy@(img-0.5)@Fx.T ]  (B,150)
// Uses identity glimpse(img-0.5) = glimpse(img) - 0.5*sumFy[i]*sumFx[j].
// One block (256 thr) per batch element.
// ---------------------------------------------------------------------------
__global__ void glimpse_k(const float* x, const float* Fx, const float* Fy,
                          const float* gamma, float* r)
{
    const int b = blockIdx.x, t = threadIdx.x;
    __shared__ float Fxs[320], Fys[320], sF[10], tb[960];
    for (int e = t; e < 320; e += 256) { Fxs[e] = Fx[b * 320 + e]; Fys[e] = Fy[b * 320 + e]; }
    __syncthreads();
    if (t < 5) {
        float s = 0.0f; for (int a = 0; a < 64; ++a) s += Fys[t * 64 + a]; sF[t] = s;
    } else if (t >= 8 && t < 13) {
        const int j = t - 8;
        float s = 0.0f; for (int a = 0; a < 64; ++a) s += Fxs[j * 64 + a]; sF[5 + j] = s;
    }
    __syncthreads();
    for (int e = t; e < 960; e += 256) {            // tb[c][i][x] = sum_y Fy[i][y]*img[c][y][x]
        const int c = e / 320, rem = e % 320, i = rem >> 6, xx = rem & 63;
        const float* img = x + ((size_t)(b * CCH + c) * 64) * 64 + xx;
        float acc = 0.0f;
        for (int y = 0; y < 64; ++y) acc += Fys[i * 64 + y] * img[y * 64];
        tb[e] = acc;
    }
    __syncthreads();
    const float gm = gamma[b];
    for (int e = t; e < 75; e += 256) {
        const int c = e / 25, ij = e % 25, i = ij / 5, j = ij % 5;
        float acc = 0.0f;
        const float* tr = &tb[c * 320 + i * 64];
        const float* fr = &Fxs[j * 64];
        for (int xx = 0; xx < 64; ++xx) acc += tr[xx] * fr[xx];
        r[b * 150 + e]      = acc * gm;
        r[b * 150 + 75 + e] = (acc - 0.5f * sF[i] * sF[5 + j]) * gm;
    }
}

// ---------------------------------------------------------------------------
// q = mu + eps * exp(sig)
// ---------------------------------------------------------------------------
__global__ void q_k(const float* mu, const float* sig, const float* eps,
                    float* q, int total)
{
    const int idx = blockIdx.x * blockDim.x + threadIdx.x;
    if (idx >= total) return;
    q[idx] = mu[idx] + eps[idx] * expf(sig[idx]);
}

// ---------------------------------------------------------------------------
// Alignment: logits[l] = w2 . tanh(u[b,l,:]+v[b,:]); softmax over l;
// sent[b,:] = sum_l prob[l]*h_lang[b,l,:].  One block (256 thr) per batch.
// ---------------------------------------------------------------------------
__global__ void align_sent_k(const float* u, const float* v, const float* w2,
                             const float* h_lang, float* sent)
{
    const int b = blockIdx.x, t = threadIdx.x;
    const int lane = t & 31, wave = t >> 5;
    __shared__ float logit[LLEN], probs[LLEN];
    for (int l = wave; l < LLEN; l += 8) {
        const float* ur = u + (size_t)(b * LLEN + l) * ALV;
        const float* vr = v + (size_t)b * ALV;
        float acc = 0.0f;
        for (int k = lane; k < ALV; k += 32)
            acc += w2[k] * tanhf(ur[k] + vr[k]);
        for (int off = 16; off; off >>= 1) acc += __shfl_xor(acc, off, 32);
        if (lane == 0) logit[l] = acc;
    }
    __syncthreads();
    if (t == 0) {
        float m = logit[0];
        for (int l = 1; l < LLEN; ++l) m = fmaxf(m, logit[l]);
        float s = 0.0f;
        for (int l = 0; l < LLEN; ++l) { const float e = expf(logit[l] - m); probs[l] = e; s += e; }
        const float inv = 1.0f / s;
        for (int l = 0; l < LLEN; ++l) probs[l] *= inv;
    }
    __syncthreads();
    for (int j = t; j < 2 * LHV; j += 256) {
        float acc = 0.0f;
        for (int l = 0; l < LLEN; ++l)
            acc += probs[l] * h_lang[(size_t)(b * LLEN + l) * (2 * LHV) + j];
        sent[b * (2 * LHV) + j] = acc;
    }
}

// ---------------------------------------------------------------------------
// Write patch: out[b,c,y,x] = (sum_{i,j} Fyw[i,y]*w[c,i,j]*Fxw[j,x]) / gw
// One block (256 thr) per batch element.
// ---------------------------------------------------------------------------
__global__ void write_out_k(const float* w_img, const float* Fxw, const float* Fyw,
                            const float* gw, float* outp)
{
    const int b = blockIdx.x, t = threadIdx.x;
    __shared__ float Fxs[320], Fys[320], wv[75], t2[960];
    for (int e = t; e < 320; e += 256) { Fxs[e] = Fxw[b * 320 + e]; Fys[e] = Fyw[b * 320 + e]; }
    for (int e = t; e < 75; e += 256) wv[e] = w_img[b * 75 + e];
    __syncthreads();
    for (int e = t; e < 960; e += 256) {            // t2[c][i][x] = sum_j w[c][i][j]*Fxw[j][x]
        const int c = e / 320, rem = e % 320, i = rem >> 6, xx = rem & 63;
        float acc = 0.0f;
        for (int j = 0; j < 5; ++j) acc += wv[c * 25 + i * 5 + j] * Fxs[j * 64 + xx];
        t2[e] = acc;
    }
    __syncthreads();
    const float invg = 1.0f / gw[b];
    float* ob = outp + (size_t)b * (CCH * IMG_A * IMG_A);
    for (int e = t; e < CCH * IMG_A * IMG_A; e += 256) {
        const int c = e >> 12, rem = e & 4095, y = rem >> 6, xx = rem & 63;
        float acc = 0.0f;
        for (int i = 0; i < 5; ++i) acc += Fys[i * 64 + y] * t2[c * 320 + i * 64 + xx];
        ob[e] = acc * invg;
    }
}

// ---------------------------------------------------------------------------
// Host side
// ---------------------------------------------------------------------------
static inline GemmArgs ga1(const float* a, int lda, const float* w, int ldw, int width)
{
    GemmArgs g{};
    g.s[0].a = a; g.s[0].lda = lda;
    g.s[0].w = w; g.s[0].ldw = ldw;
    g.s[0].width = width;
    g.ns = 1;
    return g;
}

static inline void launch_gemm(hipStream_t st, GemmArgs G,
                               const float* b1, const float* b2,
                               const float* cadd, int ldca,
                               float* out, int ldc, int M, int N)
{
    const int colsPerBlock = 4 * 16 * NTILE;   // 4 waves x 4 tiles x 16
    dim3 grid((unsigned)((N + colsPerBlock - 1) / colsPerBlock),
              (unsigned)((M + 15) / 16));
    gemm_wmma_f32<<<grid, 128, 0, st>>>(G, b1, b2, cadd, ldca, out, ldc, M, N);
}

extern "C" void kernel_launch(void* const* d_in, const int* in_sizes, int n_in,
                              void* d_out, int out_size, void* d_ws, size_t ws_size,
                              hipStream_t stream)
{
    (void)in_sizes; (void)n_in; (void)out_size; (void)ws_size;

    const float* x        = (const float*)d_in[0];
    const float* y        = (const float*)d_in[1];
    const float* eps_q    = (const float*)d_in[2];
    const float* enc_Wih  = (const float*)d_in[3];
    const float* enc_Whh  = (const float*)d_in[4];
    const float* enc_bih  = (const float*)d_in[5];
    const float* enc_bhh  = (const float*)d_in[6];
    const float* dec_Wih  = (const float*)d_in[7];
    const float* dec_Whh  = (const float*)d_in[8];
    const float* dec_bih  = (const float*)d_in[9];
    const float* dec_bhh  = (const float*)d_in[10];
    const float* mu_W     = (const float*)d_in[11];
    const float* mu_b     = (const float*)d_in[12];
    const float* sig_W    = (const float*)d_in[13];
    const float* sig_b    = (const float*)d_in[14];
    const float* write_W  = (const float*)d_in[15];
    const float* write_b  = (const float*)d_in[16];
    const float* align_W1 = (const float*)d_in[17];
    const float* align_b1 = (const float*)d_in[18];
    const float* align_w2 = (const float*)d_in[19];
    const float* attn_W   = (const float*)d_in[20];
    const float* attn_b   = (const float*)d_in[21];
    const float* lf_Wih   = (const float*)d_in[22];
    const float* lf_Whh   = (const float*)d_in[23];
    const float* lf_bih   = (const float*)d_in[24];
    const float* lf_bhh   = (const float*)d_in[25];
    const float* lb_Wih   = (const float*)d_in[26];
    const float* lb_Whh   = (const float*)d_in[27];
    const float* lb_bih   = (const float*)d_in[28];
    const float* lb_bhh   = (const float*)d_in[29];

    float* out = (float*)d_out;

    // ---- workspace bump allocator ----
    float* wsf = (float*)d_ws;
    size_t off = 0;
    auto alloc = [&](size_t n) -> float* {
        float* p = wsf + off;
        off += (n + 63) & ~(size_t)63;
        return p;
    };
    float* pre_f   = alloc((size_t)BN * LLEN * 4 * LHV);   // 4096 x 512
    float* pre_b   = alloc((size_t)BN * LLEN * 4 * LHV);
    float* h_lang  = alloc((size_t)BN * LLEN * 2 * LHV);   // (B,L,256)
    float* u_buf   = alloc((size_t)BN * LLEN * ALV);       // 4096 x 512
    float* h_lf    = alloc((size_t)BN * LHV);
    float* c_lf    = alloc((size_t)BN * LHV);
    float* h_lb    = alloc((size_t)BN * LHV);
    float* c_lb    = alloc((size_t)BN * LHV);
    float* gates_l = alloc((size_t)BN * 4 * LHV);
    float* h_enc   = alloc((size_t)BN * ENCV);
    float* c_enc   = alloc((size_t)BN * ENCV);
    float* h_dec   = alloc((size_t)BN * DECV);
    float* c_dec   = alloc((size_t)BN * DECV);
    float* gates_e = alloc((size_t)BN * 4 * ENCV);
    float* gates_d = alloc((size_t)BN * 4 * DECV);
    float* Fx      = alloc((size_t)BN * 5 * 64);
    float* Fy      = alloc((size_t)BN * 5 * 64);
    float* gam     = alloc((size_t)BN);
    float* Fxw     = alloc((size_t)BN * 5 * 64);
    float* Fyw     = alloc((size_t)BN * 5 * 64);
    float* gw      = alloc((size_t)BN);
    float* r_buf   = alloc((size_t)BN * 150);
    float* mu_buf  = alloc((size_t)BN * ZV);
    float* sig_buf = alloc((size_t)BN * ZV);
    float* q_buf   = alloc((size_t)BN * ZV);
    float* v_buf   = alloc((size_t)BN * ALV);
    float* sent    = alloc((size_t)BN * 2 * LHV);
    float* w_img   = alloc((size_t)BN * 75);

    // ---- zero-init recurrent state ----
    hipMemsetAsync(h_lf, 0, sizeof(float) * BN * LHV, stream);
    hipMemsetAsync(c_lf, 0, sizeof(float) * BN * LHV, stream);
    hipMemsetAsync(h_lb, 0, sizeof(float) * BN * LHV, stream);
    hipMemsetAsync(c_lb, 0, sizeof(float) * BN * LHV, stream);
    hipMemsetAsync(h_enc, 0, sizeof(float) * BN * ENCV, stream);
    hipMemsetAsync(c_enc, 0, sizeof(float) * BN * ENCV, stream);
    hipMemsetAsync(h_dec, 0, sizeof(float) * BN * DECV, stream);
    hipMemsetAsync(c_dec, 0, sizeof(float) * BN * DECV, stream);

    // ---- bi-LSTM: hoist the input GEMMs (x @ Wih^T + biases) out of the scan ----
    launch_gemm(stream, ga1(y, LINV, lf_Wih, LINV, LINV),
                lf_bih, lf_bhh, nullptr, 0, pre_f, 4 * LHV, BN * LLEN, 4 * LHV);
    launch_gemm(stream, ga1(y, LINV, lb_Wih, LINV, LINV),
                lb_bih, lb_bhh, nullptr, 0, pre_b, 4 * LHV, BN * LLEN, 4 * LHV);

    for (int s = 0; s < LLEN; ++s) {
        // forward step l = s
        launch_gemm(stream, ga1(h_lf, LHV, lf_Whh, LHV, LHV),
                    nullptr, nullptr, pre_f + (size_t)s * 4 * LHV, LLEN * 4 * LHV,
                    gates_l, 4 * LHV, BN, 4 * LHV);
        lstm_cell_k<<<(BN * LHV + 255) / 256, 256, 0, stream>>>(
            gates_l, c_lf, h_lf, c_lf, LHV,
            h_lang + (size_t)s * 2 * LHV, LLEN * 2 * LHV, BN * LHV);
        // backward step l = LLEN-1-s
        const int l = LLEN - 1 - s;
        launch_gemm(stream, ga1(h_lb, LHV, lb_Whh, LHV, LHV),
                    nullptr, nullptr, pre_b + (size_t)l * 4 * LHV, LLEN * 4 * LHV,
                    gates_l, 4 * LHV, BN, 4 * LHV);
        lstm_cell_k<<<(BN * LHV + 255) / 256, 256, 0, stream>>>(
            gates_l, c_lb, h_lb, c_lb, LHV,
            h_lang + (size_t)l * 2 * LHV + LHV, LLEN * 2 * LHV, BN * LHV);
    }

    // ---- time-invariant half of the alignment: u = h_lang @ W1[:,512:768]^T ----
    launch_gemm(stream,
                ga1(h_lang, 2 * LHV, align_W1 + DECV, 2 * LHV + DECV, 2 * LHV),
                nullptr, nullptr, nullptr, 0, u_buf, ALV, BN * LLEN, ALV);

    // ---- DRAW loop ----
    for (int t = 0; t < TSTEPS; ++t) {
        attn_k<<<BN, 64, 0, stream>>>(h_dec, attn_W, attn_b, Fx, Fy, gam, RN5);
        glimpse_k<<<BN, 256, 0, stream>>>(x, Fx, Fy, gam, r_buf);

        // encoder gates: [r | h_dec | h_enc] @ [enc_Wih | enc_Whh]^T + biases
        {
            GemmArgs ge{};
            ge.ns = 3;
            ge.s[0] = { r_buf, 150, enc_Wih,       662, 150 };
            ge.s[1] = { h_dec, DECV, enc_Wih + 150, 662, DECV };
            ge.s[2] = { h_enc, ENCV, enc_Whh,      ENCV, ENCV };
            launch_gemm(stream, ge, enc_bih, enc_bhh, nullptr, 0,
                        gates_e, 4 * ENCV, BN, 4 * ENCV);
        }
        lstm_cell_k<<<(BN * ENCV + 255) / 256, 256, 0, stream>>>(
            gates_e, c_enc, h_enc, c_enc, ENCV, nullptr, 0, BN * ENCV);

        launch_gemm(stream, ga1(h_enc, ENCV, mu_W, ENCV, ENCV),
                    mu_b, nullptr, nullptr, 0, mu_buf, ZV, BN, ZV);
        launch_gemm(stream, ga1(h_enc, ENCV, sig_W, ENCV, ENCV),
                    sig_b, nullptr, nullptr, 0, sig_buf, ZV, BN, ZV);
        q_k<<<(BN * ZV + 255) / 256, 256, 0, stream>>>(
            mu_buf, sig_buf, eps_q + (size_t)t * BN * ZV, q_buf, BN * ZV);

        // per-step half of alignment: v = h_dec @ W1[:,0:512]^T + b1
        launch_gemm(stream, ga1(h_dec, DECV, align_W1, 2 * LHV + DECV, DECV),
                    align_b1, nullptr, nullptr, 0, v_buf, ALV, BN, ALV);
        align_sent_k<<<BN, 256, 0, stream>>>(u_buf, v_buf, align_w2, h_lang, sent);

        // decoder gates: [q | sent | h_dec] @ [dec_Wih | dec_Whh]^T + biases
        {
            GemmArgs gd{};
            gd.ns = 3;
            gd.s[0] = { q_buf, ZV,       dec_Wih,      356, ZV };
            gd.s[1] = { sent,  2 * LHV,  dec_Wih + ZV, 356, 2 * LHV };
            gd.s[2] = { h_dec, DECV,     dec_Whh,      DECV, DECV };
            launch_gemm(stream, gd, dec_bih, dec_bhh, nullptr, 0,
                        gates_d, 4 * DECV, BN, 4 * DECV);
        }
        lstm_cell_k<<<(BN * DECV + 255) / 256, 256, 0, stream>>>(
            gates_d, c_dec, h_dec, c_dec, DECV, nullptr, 0, BN * DECV);

        // write patch + write attention from updated h_dec
        launch_gemm(stream, ga1(h_dec, DECV, write_W, DECV, DECV),
                    write_b, nullptr, nullptr, 0, w_img, 75, BN, 75);
        attn_k<<<BN, 64, 0, stream>>>(h_dec, attn_W, attn_b, Fxw, Fyw, gw, RN5);
        write_out_k<<<BN, 256, 0, stream>>>(
            w_img, Fxw, Fyw, gw,
            out + (size_t)t * BN * CCH * IMG_A * IMG_A);
    }
}